// DecisionTree_49005576847578
// MI455X (gfx1250) — compile-verified
//
#include <hip/hip_runtime.h>

typedef float v2f __attribute__((ext_vector_type(2)));
typedef float v4f __attribute__((ext_vector_type(4)));
typedef float v8f __attribute__((ext_vector_type(8)));

// One block per row (B=4096). 256 threads = 8 waves (wave32).
// out[r, j] with j = a*1024 + b*64 + t,  a,b in [0,16), t in [0,64)
//   = p_a[a] * p_b[b] * tail[t]
// M[a,b] = p_a[a]*p_b[b] computed with one V_WMMA_F32_16X16X4_F32 rank-1 product.
__global__ void __launch_bounds__(256)
decision_tree_kron_kernel(const float* __restrict__ x,
                          const float* __restrict__ cp,
                          float* __restrict__ out) {
    const int r   = blockIdx.x;
    const int tid = threadIdx.x;

    __shared__ float s_bins[7][4];
    __shared__ float s_pa[16];
    __shared__ float s_pb[16];
    __shared__ float s_tail[64];
    __shared__ float s_M[256];

    // ---- Stage 1: per-feature 4-way softmax (threads 0..6) ----
    if (tid < 7) {
        const int f = tid;
        const float xv = x[(size_t)r * 7 + f];
        const float c0 = cp[f * 3 + 0];
        const float c1 = cp[f * 3 + 1];
        const float c2 = cp[f * 3 + 2];
        // sort 3 cutpoints ascending
        const float lo  = fminf(fminf(c0, c1), c2);
        const float hi  = fmaxf(fmaxf(c0, c1), c2);
        const float mid = (c0 + c1 + c2) - lo - hi;
        // b = cumsum([0, -lo, -mid, -hi])
        const float b0 = 0.0f;
        const float b1 = -lo;
        const float b2 = b1 - mid;
        const float b3 = b2 - hi;
        // h_k = x*(k+1) + b_k ; logits = h/T with T = 0.1
        const float t0 = (xv * 1.0f + b0) * 10.0f;
        const float t1 = (xv * 2.0f + b1) * 10.0f;
        const float t2 = (xv * 3.0f + b2) * 10.0f;
        const float t3 = (xv * 4.0f + b3) * 10.0f;
        const float m  = fmaxf(fmaxf(t0, t1), fmaxf(t2, t3));
        const float e0 = __builtin_expf(t0 - m);
        const float e1 = __builtin_expf(t1 - m);
        const float e2 = __builtin_expf(t2 - m);
        const float e3 = __builtin_expf(t3 - m);
        const float inv = 1.0f / (e0 + e1 + e2 + e3);
        s_bins[f][0] = e0 * inv;
        s_bins[f][1] = e1 * inv;
        s_bins[f][2] = e2 * inv;
        s_bins[f][3] = e3 * inv;
    }
    __syncthreads();

    // ---- Stage 2: partial Kronecker factors ----
    if (tid < 16) {
        s_pa[tid] = s_bins[0][tid >> 2] * s_bins[1][tid & 3];
    } else if (tid < 32) {
        const int i = tid - 16;
        s_pb[i] = s_bins[2][i >> 2] * s_bins[3][i & 3];
    } else if (tid < 96) {
        const int t = tid - 32;
        s_tail[t] = s_bins[4][t >> 4] * s_bins[5][(t >> 2) & 3] * s_bins[6][t & 3];
    }
    __syncthreads();

    // ---- Stage 3: 16x16 outer product via WMMA (wave 0, EXEC all ones) ----
    if (tid < 32) {
        const int lane = tid;
        // A (16x4 f32): lanes 0-15 hold K=0 (VGPR0) / K=1 (VGPR1); lanes 16-31 hold K=2/K=3.
        // Only K=0 column is non-zero: A[m][0] = p_a[m].
        v2f a;
        a.x = (lane < 16) ? s_pa[lane] : 0.0f;
        a.y = 0.0f;
        // B (4x16 f32): rows striped across lanes; only K=0 row non-zero: B[0][n] = p_b[n].
        v2f b;
        b.x = (lane < 16) ? s_pb[lane] : 0.0f;
        b.y = 0.0f;
        v8f c = {};
        v8f d = __builtin_amdgcn_wmma_f32_16x16x4_f32(
            /*neg_a=*/false, a, /*neg_b=*/false, b,
            /*c_mod=*/(short)0, c, /*reuse_a=*/false, /*reuse_b=*/false);
        // D layout: VGPR i -> M=i (lanes 0-15) or M=i+8 (lanes 16-31); N = lane&15.
        const int ncol = lane & 15;
        const int mbase = (lane < 16) ? 0 : 8;
#pragma unroll
        for (int i = 0; i < 8; ++i) {
            s_M[(mbase + i) * 16 + ncol] = d[i];
        }
    }
    __syncthreads();

    // ---- Stage 4: stream 16384 outputs with coalesced non-temporal float4 stores ----
    const int tb = (tid * 4) & 63;
    const float t0 = s_tail[tb + 0];
    const float t1 = s_tail[tb + 1];
    const float t2 = s_tail[tb + 2];
    const float t3 = s_tail[tb + 3];
    float* const row_out = out + (size_t)r * 16384;
#pragma unroll
    for (int cchunk = 0; cchunk < 16; ++cchunk) {
        const int j = cchunk * 1024 + tid * 4;   // 16B aligned
        const float m = s_M[j >> 6];
        v4f v;
        v.x = m * t0;
        v.y = m * t1;
        v.z = m * t2;
        v.w = m * t3;
        __builtin_nontemporal_store(v, (v4f*)(row_out + j));
    }
}

extern "C" void kernel_launch(void* const* d_in, const int* in_sizes, int n_in,
                              void* d_out, int out_size, void* d_ws, size_t ws_size,
                              hipStream_t stream) {
    const float* x  = (const float*)d_in[0];   // (B, 7) fp32
    const float* cp = (const float*)d_in[1];   // (7, 3) fp32
    float* out      = (float*)d_out;           // (B, 16384) fp32
    const int B = in_sizes[0] / 7;
    decision_tree_kron_kernel<<<dim3(B), dim3(256), 0, stream>>>(x, cp, out);
}